// MultiHeadAttention_SMC_22557168239003
// MI455X (gfx1250) — compile-verified
//
#include <hip/hip_runtime.h>

#define B_ 8
#define P_ 16
#define S_ 512
#define D_ 1024
#define H_ 16
#define DEPTH_ 64
#define SIGMA_ 0.1f
#define BP_ (B_ * P_)            // 128 (b,p) units

typedef __attribute__((ext_vector_type(2))) float v2f;
typedef __attribute__((ext_vector_type(4))) float f4;
typedef __attribute__((ext_vector_type(8))) float v8f;

// ---------------------------------------------------------------------------
// Projection GEMM:  Y[m][n] = sum_k X[m][k]*W[k][n] + bias[n] + SIGMA*eps[m][n]
// M = 128, N = K = 1024, fp32 via V_WMMA_F32_16X16X4_F32.
// 8 waves / block, one 16x16 output tile per wave, 512 tiles -> 64 blocks.
//
// A layout (ISA 7.12.2, 32-bit A 16x4): lanes 0-15 -> M rows, VGPR0/1 = K0/K1;
// lanes 16-31 -> same M rows, VGPR0/1 = K2/K3.
// B layout (4x16, K-major symmetric): lane half selects K pair, lane&15 = N.
// C/D: VGPR j -> M = (half*8 + j), N = lane&15.
// ---------------------------------------------------------------------------
__global__ __launch_bounds__(256) void proj_wmma(const float* __restrict__ X,
                                                 const float* __restrict__ W,
                                                 const float* __restrict__ bias,
                                                 const float* __restrict__ eps,
                                                 float* __restrict__ Y) {
  const int wave = threadIdx.x >> 5;
  const int lane = threadIdx.x & 31;
  const int half = lane >> 4;      // 0: K0/K1, 1: K2/K3
  const int lm   = lane & 15;      // A: row in tile, B: col in tile
  const int tile = blockIdx.x * 8 + wave;      // 0..511
  const int m0 = (tile >> 6) * 16;             // 8 M-tiles
  const int n0 = (tile & 63) * 16;             // 64 N-tiles

  v8f acc = {0.f, 0.f, 0.f, 0.f, 0.f, 0.f, 0.f, 0.f};
  const float* xrow = X + (size_t)(m0 + lm) * D_;

#pragma unroll 4
  for (int k0 = 0; k0 < D_; k0 += 4) {
    const int ka = k0 + half * 2;
    v2f a;
    a.x = xrow[ka];
    a.y = xrow[ka + 1];
    v2f b;
    b.x = W[(size_t)ka * D_ + n0 + lm];
    b.y = W[(size_t)(ka + 1) * D_ + n0 + lm];
    acc = __builtin_amdgcn_wmma_f32_16x16x4_f32(false, a, false, b,
                                                (short)0, acc, false, false);
  }

  const int n = n0 + lm;
  const float bn = bias[n];
#pragma unroll
  for (int j = 0; j < 8; ++j) {
    const int m = m0 + half * 8 + j;
    const size_t idx = (size_t)m * D_ + n;
    Y[idx] = acc[j] + bn + SIGMA_ * eps[idx];
  }
}

// ---------------------------------------------------------------------------
// Fused KV-cache write + single-pass attention for one (b,p).
// 512 threads = 16 waves. Wave w owns S-rows [32w, 32w+32).
// Pass 1: stream K -> K_out (k_ substituted at timestep) + logits.
// Softmax: wave w = head w.
// Pass 2: stream V -> V_out (v_ substituted) + context accumulation.
// ---------------------------------------------------------------------------
__global__ __launch_bounds__(512) void attn_fused(
    const float* __restrict__ Kin, const float* __restrict__ Vin,
    const float* __restrict__ qp, const float* __restrict__ kp,
    const float* __restrict__ vp, const int* __restrict__ ts_ptr,
    float* __restrict__ Kout, float* __restrict__ Vout,
    float* __restrict__ Aout, float* __restrict__ zctx) {
  __shared__ float qs[D_];
  __shared__ float ks_[D_];
  __shared__ float vs_[D_];
  __shared__ float lg[H_ * S_];   // logits, then attention weights
  __shared__ float zacc[D_];

  const int bp = blockIdx.x;      // 0..127
  const int t = threadIdx.x;
  const int w = t >> 5;
  const int lane = t & 31;
  const int ts = *ts_ptr;

  qs[t] = qp[(size_t)bp * D_ + t];
  qs[t + 512] = qp[(size_t)bp * D_ + t + 512];
  ks_[t] = kp[(size_t)bp * D_ + t];
  ks_[t + 512] = kp[(size_t)bp * D_ + t + 512];
  vs_[t] = vp[(size_t)bp * D_ + t];
  vs_[t + 512] = vp[(size_t)bp * D_ + t + 512];
  zacc[t] = 0.f;
  zacc[t + 512] = 0.f;
  __syncthreads();

  const size_t slab = (size_t)bp * S_ * D_;

  // ---- Pass 1: K stream + logits -----------------------------------------
  {
    const f4* __restrict__ Kb = (const f4*)(Kin + slab);
    f4* __restrict__ Ko = (f4*)(Kout + slab);
    const f4* ksv = (const f4*)ks_;
    for (int s = w * 32; s < w * 32 + 32; ++s) {
      float accl[8];
#pragma unroll
      for (int i = 0; i < 8; ++i) {
        const int f4i = i * 32 + lane;          // 0..255 within row
        f4 val = (s == ts) ? ksv[f4i]
                           : __builtin_nontemporal_load(Kb + (size_t)s * 256 + f4i);
        __builtin_nontemporal_store(val, Ko + (size_t)s * 256 + f4i);
        const int c = f4i * 4;
        accl[i] = val.x * qs[c] + val.y * qs[c + 1] +
                  val.z * qs[c + 2] + val.w * qs[c + 3];
      }
#pragma unroll
      for (int i = 0; i < 8; ++i) {
        float vsum = accl[i];                    // half-wave reduce (16 lanes)
        vsum += __shfl_xor(vsum, 8, 16);
        vsum += __shfl_xor(vsum, 4, 16);
        vsum += __shfl_xor(vsum, 2, 16);
        vsum += __shfl_xor(vsum, 1, 16);
        if ((lane & 15) == 0) {
          const int h = 2 * i + (lane >> 4);     // lanes 0-15: head 2i, 16-31: 2i+1
          lg[h * S_ + s] = vsum * 0.125f;        // 1/sqrt(64)
        }
      }
    }
  }
  __syncthreads();

  // ---- Softmax: wave w handles head w ------------------------------------
  {
    const int h = w;
    float vals[16];
    float m = -INFINITY;
#pragma unroll
    for (int j = 0; j < 16; ++j) {
      vals[j] = lg[h * S_ + lane + j * 32];
      m = fmaxf(m, vals[j]);
    }
    for (int off = 16; off >= 1; off >>= 1) m = fmaxf(m, __shfl_xor(m, off, 32));
    float sum = 0.f;
#pragma unroll
    for (int j = 0; j < 16; ++j) {
      vals[j] = __expf(vals[j] - m);
      sum += vals[j];
    }
    for (int off = 16; off >= 1; off >>= 1) sum += __shfl_xor(sum, off, 32);
    const float inv = 1.0f / sum;
    float* aout = Aout + ((size_t)bp * H_ + h) * S_;
#pragma unroll
    for (int j = 0; j < 16; ++j) {
      const float a = vals[j] * inv;
      lg[h * S_ + lane + j * 32] = a;            // reuse as attention weights
      aout[lane + j * 32] = a;
    }
  }
  __syncthreads();

  // ---- Pass 2: V stream + context ----------------------------------------
  {
    const f4* __restrict__ Vb = (const f4*)(Vin + slab);
    f4* __restrict__ Vo = (f4*)(Vout + slab);
    const f4* vsv = (const f4*)vs_;
    f4 zloc[8] = {};
    for (int s = w * 32; s < w * 32 + 32; ++s) {
#pragma unroll
      for (int i = 0; i < 8; ++i) {
        const int f4i = i * 32 + lane;
        f4 val = (s == ts) ? vsv[f4i]
                           : __builtin_nontemporal_load(Vb + (size_t)s * 256 + f4i);
        __builtin_nontemporal_store(val, Vo + (size_t)s * 256 + f4i);
        const int h = f4i >> 4;                  // head owning these 4 cols
        const float a = lg[h * S_ + s];
        zloc[i].x += a * val.x;
        zloc[i].y += a * val.y;
        zloc[i].z += a * val.z;
        zloc[i].w += a * val.w;
      }
    }
#pragma unroll
    for (int i = 0; i < 8; ++i) {
      const int c = (i * 32 + lane) * 4;
      atomicAdd(&zacc[c + 0], zloc[i].x);        // ds_add_f32, conflict-free per wave
      atomicAdd(&zacc[c + 1], zloc[i].y);
      atomicAdd(&zacc[c + 2], zloc[i].z);
      atomicAdd(&zacc[c + 3], zloc[i].w);
    }
  }
  __syncthreads();

  zctx[(size_t)bp * D_ + t] = zacc[t];
  zctx[(size_t)bp * D_ + t + 512] = zacc[t + 512];
}

// ---------------------------------------------------------------------------
extern "C" void kernel_launch(void* const* d_in, const int* in_sizes, int n_in,
                              void* d_out, int out_size, void* d_ws, size_t ws_size,
                              hipStream_t stream) {
  const float* q     = (const float*)d_in[0];
  const float* k     = (const float*)d_in[1];
  const float* v     = (const float*)d_in[2];
  const float* Kin   = (const float*)d_in[3];
  const float* Vin   = (const float*)d_in[4];
  const float* Wq    = (const float*)d_in[5];
  const float* bq    = (const float*)d_in[6];
  const float* Wk    = (const float*)d_in[7];
  const float* bk    = (const float*)d_in[8];
  const float* Wv    = (const float*)d_in[9];
  const float* bv    = (const float*)d_in[10];
  const float* Wo    = (const float*)d_in[11];
  const float* bo    = (const float*)d_in[12];
  const float* eps_q = (const float*)d_in[13];
  const float* eps_k = (const float*)d_in[14];
  const float* eps_v = (const float*)d_in[15];
  const float* eps_z = (const float*)d_in[16];
  const int*   ts    = (const int*)d_in[17];

  float* out = (float*)d_out;
  float* z_out = out;                                    // (B,P,1,D)  = 131072
  float* K_out = out + (size_t)BP_ * D_;                 // (B,P,S,D)  = 64M
  float* V_out = K_out + (size_t)BP_ * S_ * D_;
  float* A_out = V_out + (size_t)BP_ * S_ * D_;          // (B,P,H,1,S) = 1M

  float* ws = (float*)d_ws;
  float* qp = ws;                                        // 128x1024 each
  float* kp = ws + (size_t)BP_ * D_;
  float* vp = ws + 2 * (size_t)BP_ * D_;
  float* zc = ws + 3 * (size_t)BP_ * D_;

  // q/k/v projections + sigma noise (fp32 WMMA)
  proj_wmma<<<64, 256, 0, stream>>>(q, Wq, bq, eps_q, qp);
  proj_wmma<<<64, 256, 0, stream>>>(k, Wk, bk, eps_k, kp);
  proj_wmma<<<64, 256, 0, stream>>>(v, Wv, bv, eps_v, vp);

  // fused KV-cache write + attention (single pass over K and V)
  attn_fused<<<BP_, 512, 0, stream>>>(Kin, Vin, qp, kp, vp, ts,
                                      K_out, V_out, A_out, zc);

  // output projection + sigma noise -> z
  proj_wmma<<<64, 256, 0, stream>>>(zc, Wo, bo, eps_z, z_out);
}